// L1Loss_6820408066429
// MI455X (gfx1250) — compile-verified
//
#include <hip/hip_runtime.h>

typedef __attribute__((ext_vector_type(2))) float v2f;
typedef __attribute__((ext_vector_type(4))) float v4f;
typedef __attribute__((ext_vector_type(8))) float v8f;

// ---------------------------------------------------------------------------
// Wave-level sum of one float per lane (wave32) using V_WMMA_F32_16X16X4_F32.
// A (16x4 f32): lanes 0-15 -> A[M][K=0], lanes 16-31 -> A[M][K=2]; K=1,3 zero.
// B (4x16) = ones.  D[M][N] = acc(M) + acc(M+16); summing D's 8 VGPRs plus one
// xor-16 shuffle-add yields the exact fp32 32-lane sum in every lane.
// Requires EXEC all ones (call only from fully-active 256-thread blocks).
// ---------------------------------------------------------------------------
__device__ __forceinline__ float wave_sum32(float acc) {
  v2f a; a[0] = acc;  a[1] = 0.0f;
  v2f b; b[0] = 1.0f; b[1] = 1.0f;
  v8f c = {};
  v8f d = __builtin_amdgcn_wmma_f32_16x16x4_f32(
      /*neg_a=*/false, a, /*neg_b=*/false, b,
      /*c_mod=*/(short)0, c, /*reuse_a=*/false, /*reuse_b=*/false);
  float s = ((d[0] + d[1]) + (d[2] + d[3])) + ((d[4] + d[5]) + (d[6] + d[7]));
  s += __shfl_xor(s, 16, 32);
  return s;
}

// Block-level deterministic sum (blockDim.x == 256 -> 8 wave32 waves).
__device__ __forceinline__ float block_sum(float acc, float* lds) {
  float w = wave_sum32(acc);
  const int lane = threadIdx.x & 31;
  const int wid  = threadIdx.x >> 5;
  if (lane == 0) lds[wid] = w;
  __syncthreads();
  const int nw = blockDim.x >> 5;
  float total = 0.0f;
  for (int i = 0; i < nw; ++i) total += lds[i];  // uniform, deterministic
  return total;
}

// ---------------------------------------------------------------------------
// Pass 1: streaming |a-b| with 4x-unrolled 128-bit non-temporal loads
// (8 global_load_b128 issued back-to-back per iteration for MLP).
// ---------------------------------------------------------------------------
__global__ void __launch_bounds__(256)
l1_partial_kernel(const float* __restrict__ yhat, const float* __restrict__ y,
                  float* __restrict__ partial, long long n4, long long n) {
  __shared__ float lds[16];
  const long long tid    = (long long)blockIdx.x * blockDim.x + threadIdx.x;
  const long long stride = (long long)gridDim.x * blockDim.x;

  const v4f* a4 = (const v4f*)yhat;
  const v4f* b4 = (const v4f*)y;

  float acc0 = 0.0f, acc1 = 0.0f, acc2 = 0.0f, acc3 = 0.0f;
  long long i = tid;

  // main loop: 4 vec4s per thread per pass, spaced by stride
  for (; i + 3 * stride < n4; i += 4 * stride) {
    v4f a0 = __builtin_nontemporal_load(&a4[i]);
    v4f a1 = __builtin_nontemporal_load(&a4[i + stride]);
    v4f a2 = __builtin_nontemporal_load(&a4[i + 2 * stride]);
    v4f a3 = __builtin_nontemporal_load(&a4[i + 3 * stride]);
    v4f b0 = __builtin_nontemporal_load(&b4[i]);
    v4f b1 = __builtin_nontemporal_load(&b4[i + stride]);
    v4f b2 = __builtin_nontemporal_load(&b4[i + 2 * stride]);
    v4f b3 = __builtin_nontemporal_load(&b4[i + 3 * stride]);
    acc0 += (fabsf(a0[0] - b0[0]) + fabsf(a0[1] - b0[1])) +
            (fabsf(a0[2] - b0[2]) + fabsf(a0[3] - b0[3]));
    acc1 += (fabsf(a1[0] - b1[0]) + fabsf(a1[1] - b1[1])) +
            (fabsf(a1[2] - b1[2]) + fabsf(a1[3] - b1[3]));
    acc2 += (fabsf(a2[0] - b2[0]) + fabsf(a2[1] - b2[1])) +
            (fabsf(a2[2] - b2[2]) + fabsf(a2[3] - b2[3]));
    acc3 += (fabsf(a3[0] - b3[0]) + fabsf(a3[1] - b3[1])) +
            (fabsf(a3[2] - b3[2]) + fabsf(a3[3] - b3[3]));
  }
  // remainder vec4s
  for (; i < n4; i += stride) {
    v4f a = __builtin_nontemporal_load(&a4[i]);
    v4f b = __builtin_nontemporal_load(&b4[i]);
    acc0 += (fabsf(a[0] - b[0]) + fabsf(a[1] - b[1])) +
            (fabsf(a[2] - b[2]) + fabsf(a[3] - b[3]));
  }
  float acc = (acc0 + acc1) + (acc2 + acc3);

  // scalar tail (n not divisible by 4), handled once, deterministically
  if (blockIdx.x == 0 && threadIdx.x == 0) {
    for (long long k = n4 * 4; k < n; ++k) acc += fabsf(yhat[k] - y[k]);
  }

  float total = block_sum(acc, lds);
  if (threadIdx.x == 0) partial[blockIdx.x] = total;
}

// ---------------------------------------------------------------------------
// Pass 2: single block reduces the per-block partials and writes the mean.
// ---------------------------------------------------------------------------
__global__ void __launch_bounds__(256)
l1_final_kernel(const float* __restrict__ partial, int nparts,
                float* __restrict__ out, float inv_n) {
  __shared__ float lds[16];
  float acc = 0.0f;
  for (int i = threadIdx.x; i < nparts; i += blockDim.x) acc += partial[i];
  float total = block_sum(acc, lds);
  if (threadIdx.x == 0) out[0] = total * inv_n;
}

extern "C" void kernel_launch(void* const* d_in, const int* in_sizes, int n_in,
                              void* d_out, int out_size, void* d_ws, size_t ws_size,
                              hipStream_t stream) {
  const float* yhat = (const float*)d_in[0];
  const float* y    = (const float*)d_in[1];
  float* out        = (float*)d_out;
  float* partial    = (float*)d_ws;

  const long long n  = (long long)in_sizes[0];   // 64*128*4096 = 33,554,432
  const long long n4 = n >> 2;

  const int block = 256;   // 8 wave32 waves
  const int grid  = 2048;  // 8 KB of partials in d_ws; 16 vec4s/thread -> 4 unrolled iters

  l1_partial_kernel<<<grid, block, 0, stream>>>(yhat, y, partial, n4, n);
  l1_final_kernel<<<1, block, 0, stream>>>(partial, grid, out, 1.0f / (float)n);
}